// Upsample_1434519077113
// MI455X (gfx1250) — compile-verified
//
#include <hip/hip_runtime.h>
#include <stdint.h>

// ---- gfx1250 async global->LDS path detection -------------------------------
#if defined(__has_builtin)
#  if __has_builtin(__builtin_amdgcn_global_load_async_to_lds_b128)
#    define UF_HAVE_ASYNC_LDS 1
#  endif
#  if __has_builtin(__builtin_amdgcn_s_wait_asynccnt)
#    define UF_HAVE_WAIT_ASYNC 1
#  endif
#endif

typedef float v4f __attribute__((ext_vector_type(4)));
typedef int   v4i __attribute__((vector_size(16)));   // matches builtin param type

#define H    64
#define W    64
#define OH   128
#define OW   128
#define LDS_W 72   // floats per LDS row; interior starts at col 4 -> 16B aligned b128 targets
#define LDS_H 66   // 64 rows + 1 halo row top/bottom

__device__ __forceinline__ void g2lds_b128(const float* __restrict__ g, float* l) {
#if defined(UF_HAVE_ASYNC_LDS)
  // async copy 16B global -> LDS per lane (ASYNCcnt path, no VGPR round-trip)
  // signature (from clang diagnostic): (global v4i*, lds v4i*, imm offset, imm cpol)
  __builtin_amdgcn_global_load_async_to_lds_b128(
      (__attribute__((address_space(1))) v4i*)(unsigned long long)(const void*)g,
      (__attribute__((address_space(3))) v4i*)(unsigned)(unsigned long long)(void*)l,
      0, 0);
#else
  *(v4f*)l = *(const v4f*)g;
#endif
}

__device__ __forceinline__ void wait_async_lds() {
#if defined(UF_HAVE_ASYNC_LDS)
#  if defined(UF_HAVE_WAIT_ASYNC)
  __builtin_amdgcn_s_wait_asynccnt(0);
#  else
  asm volatile("s_wait_asynccnt 0" ::: "memory");
#  endif
#endif
}

// One block per (n,c) image: stage 64x64 input (+zero halo) in LDS once,
// emit the 128x128 upsampled output with 2x2-tap FIR per phase.
__global__ __launch_bounds__(256) void upfir2x_kernel(const float* __restrict__ x,
                                                      const float* __restrict__ kern,
                                                      float* __restrict__ out) {
  // 16B alignment is required: async b128 LDS writes and the col-4 interior
  // layout both assume the tile base sits on a 16B LDS boundary.
  __shared__ __align__(16) float tile[LDS_H * LDS_W];
  const int tid = threadIdx.x;
  const int img = blockIdx.x;
  const float* src = x + (size_t)img * (H * W);

  // 16 FIR weights, uniform across the wave -> scalar loads (s_load_b512)
  float K[16];
#pragma unroll
  for (int i = 0; i < 16; ++i) K[i] = kern[i];

  // Zero the 1-wide halo ring (out-of-image taps are zero).
  // i in [0,132): top/bottom rows; i in [132,260): left/right columns.
  for (int i = tid; i < 2 * (W + 2) + 2 * H; i += 256) {
    int y, xx;
    if (i < 2 * (W + 2)) { y = (i & 1) ? H : -1; xx = (i >> 1) - 1; }
    else { const int j = i - 2 * (W + 2); y = j >> 1; xx = (j & 1) ? W : -1; }
    tile[(y + 1) * LDS_W + 4 + xx] = 0.0f;
  }

  // Stage the 64x64 interior: 1024 b128 transfers, 4 per thread, all 16B aligned.
#pragma unroll
  for (int p = 0; p < 4; ++p) {
    const int row = p * 16 + (tid >> 4);
    const int q   = tid & 15;
    g2lds_b128(src + row * W + q * 4, &tile[(row + 1) * LDS_W + 4 + q * 4]);
  }
  wait_async_lds();
  __syncthreads();

  // Each thread: input col pair (ix, ix+1) at rows iy -> 2x4 output tile.
  const int cx = tid & 31;   // ix = 2*cx
  const int ry = tid >> 5;   // 0..7
  const int ix = 2 * cx;
  float* obase = out + (size_t)img * (OH * OW) + 4 * cx;

#pragma unroll
  for (int r = 0; r < 8; ++r) {
    const int iy = r * 8 + ry;
    // LDS rows iy-1, iy, iy+1 ; cols ix-1 .. ix+2 (halo-shifted indices)
    const float* t0 = &tile[(iy + 0) * LDS_W + 3 + ix];
    const float* t1 = t0 + LDS_W;
    const float* t2 = t1 + LDS_W;
    const float a0 = t0[0], a1 = t0[1], a2 = t0[2], a3 = t0[3];
    const float b0 = t1[0], b1 = t1[1], b2 = t1[2], b3 = t1[3];
    const float c0 = t2[0], c1 = t2[1], c2 = t2[2], c3 = t2[3];

    // phase (0,0): K[3][3],K[3][1] / K[1][3],K[1][1]; phase (0,1): K[3][2],K[3][0] / K[1][2],K[1][0]
    // phase (1,0): K[2][3],K[2][1] / K[0][3],K[0][1]; phase (1,1): K[2][2],K[2][0] / K[0][2],K[0][0]
    v4f r0, r1;
    r0.x = K[15]*a0 + K[13]*a1 + K[7]*b0 + K[5]*b1;
    r0.y = K[14]*a1 + K[12]*a2 + K[6]*b1 + K[4]*b2;
    r0.z = K[15]*a1 + K[13]*a2 + K[7]*b1 + K[5]*b2;
    r0.w = K[14]*a2 + K[12]*a3 + K[6]*b2 + K[4]*b3;
    r1.x = K[11]*b0 + K[ 9]*b1 + K[3]*c0 + K[1]*c1;
    r1.y = K[10]*b1 + K[ 8]*b2 + K[2]*c1 + K[0]*c2;
    r1.z = K[11]*b1 + K[ 9]*b2 + K[3]*c1 + K[1]*c2;
    r1.w = K[10]*b2 + K[ 8]*b3 + K[2]*c2 + K[0]*c3;

    float* prow = obase + (size_t)(2 * iy) * OW;
#if defined(__has_builtin) && __has_builtin(__builtin_nontemporal_store)
    __builtin_nontemporal_store(r0, (v4f*)prow);          // streaming: TH=NT
    __builtin_nontemporal_store(r1, (v4f*)(prow + OW));
#else
    *(v4f*)prow = r0;
    *(v4f*)(prow + OW) = r1;
#endif
  }
}

extern "C" void kernel_launch(void* const* d_in, const int* in_sizes, int n_in,
                              void* d_out, int out_size, void* d_ws, size_t ws_size,
                              hipStream_t stream) {
  const float* x    = (const float*)d_in[0];   // (8,512,64,64) f32
  const float* kern = (const float*)d_in[1];   // (4,4) f32
  float* out        = (float*)d_out;           // (8,512,128,128) f32
  const int n_img = in_sizes[0] / (H * W);     // 8*512 = 4096
  upfir2x_kernel<<<n_img, 256, 0, stream>>>(x, kern, out);
}